// AttnDecoderRNN_82995948028459
// MI455X (gfx1250) — compile-verified
//
#include <hip/hip_runtime.h>
#include <hip/hip_bf16.h>
#include <math.h>

#define H 256
#define VOCAB_FALLBACK 50257
#define LSEQ 50
#define NPART 128

typedef __attribute__((ext_vector_type(2))) float v2f;
typedef __attribute__((ext_vector_type(8))) float v8f;

// ---------------------------------------------------------------------------
// Kernel 1: whole sequential front-end in one block of 256 threads.
// ---------------------------------------------------------------------------
__device__ __forceinline__ float gru_elem(int i, const float* xv, const float* hv,
                                          const float* w_ih, const float* w_hh,
                                          const float* b_ih, const float* b_hh) {
    float gir = b_ih[i], giz = b_ih[H + i], gin = b_ih[2 * H + i];
    float ghr = b_hh[i], ghz = b_hh[H + i], ghn = b_hh[2 * H + i];
    const float* wr = w_ih + (size_t)i * H;
    const float* wz = w_ih + (size_t)(H + i) * H;
    const float* wn = w_ih + (size_t)(2 * H + i) * H;
    const float* ur = w_hh + (size_t)i * H;
    const float* uz = w_hh + (size_t)(H + i) * H;
    const float* un = w_hh + (size_t)(2 * H + i) * H;
    for (int j = 0; j < H; ++j) {
        float xj = xv[j], hj = hv[j];
        gir += wr[j] * xj; giz += wz[j] * xj; gin += wn[j] * xj;
        ghr += ur[j] * hj; ghz += uz[j] * hj; ghn += un[j] * hj;
    }
    float r = 1.f / (1.f + expf(-(gir + ghr)));
    float z = 1.f / (1.f + expf(-(giz + ghz)));
    float n = tanhf(gin + r * ghn);
    return (1.f - z) * n + z * hv[i];
}

__global__ void decoder_small(
    const int* __restrict__ token, const float* __restrict__ hidden,
    const float* __restrict__ enc, const float* __restrict__ emb,
    const float* __restrict__ attn_W, const float* __restrict__ attn_b,
    const float* __restrict__ comb_W, const float* __restrict__ comb_b,
    const float* __restrict__ w_ih_f, const float* __restrict__ w_hh_f,
    const float* __restrict__ b_ih_f, const float* __restrict__ b_hh_f,
    const float* __restrict__ w_ih_b, const float* __restrict__ w_hh_b,
    const float* __restrict__ b_ih_b, const float* __restrict__ b_hh_b,
    const float* __restrict__ w_ih_g, const float* __restrict__ w_hh_g,
    const float* __restrict__ b_ih_g, const float* __restrict__ b_hh_g,
    float* __restrict__ out_aw, float* __restrict__ out_h2,
    float* __restrict__ ws_h1) {
    __shared__ float e[H], h0[H], cat[2 * H], sc[64], awS[64], x[H];
    __shared__ float hf[H], hb[H], hm[H], h1[H];
    const int i = threadIdx.x;

    const int t = token[0];
    e[i]  = emb[(size_t)t * H + i];
    h0[i] = hidden[i];
    cat[i] = e[i];
    cat[H + i] = h0[i];
    __syncthreads();

    // attention scores: relu(attn_W @ [e;h0] + attn_b)
    if (i < LSEQ) {
        float s = attn_b[i];
        const float* w = attn_W + (size_t)i * 2 * H;
        for (int j = 0; j < 2 * H; ++j) s += w[j] * cat[j];
        sc[i] = fmaxf(s, 0.f);
    }
    __syncthreads();
    if (i == 0) {  // softmax over 50 entries (tiny, serial is fine)
        float mx = sc[0];
        for (int l = 1; l < LSEQ; ++l) mx = fmaxf(mx, sc[l]);
        float sum = 0.f;
        for (int l = 0; l < LSEQ; ++l) { awS[l] = expf(sc[l] - mx); sum += awS[l]; }
        float inv = 1.f / sum;
        for (int l = 0; l < LSEQ; ++l) awS[l] *= inv;
    }
    __syncthreads();
    if (i < LSEQ) out_aw[i] = awS[i];

    // applied = aw @ encoder_outputs
    float ap = 0.f;
    for (int l = 0; l < LSEQ; ++l) ap += awS[l] * enc[(size_t)l * H + i];
    cat[H + i] = ap;  // cat becomes [e ; applied]
    __syncthreads();

    // x = relu(comb_W @ cat + comb_b)
    {
        float s = comb_b[i];
        const float* w = comb_W + (size_t)i * 2 * H;
        for (int j = 0; j < 2 * H; ++j) s += w[j] * cat[j];
        x[i] = fmaxf(s, 0.f);
    }
    __syncthreads();

    // bidirectional GRU (seq_len = 1)
    hf[i] = gru_elem(i, x, h0, w_ih_f, w_hh_f, b_ih_f, b_hh_f);
    hb[i] = gru_elem(i, x, h0, w_ih_b, w_hh_b, b_ih_b, b_hh_b);
    hm[i] = 0.5f * (hf[i] + hb[i]);
    __syncthreads();

    // second GRU, 2 steps
    h1[i] = gru_elem(i, hf, hm, w_ih_g, w_hh_g, b_ih_g, b_hh_g);
    ws_h1[i] = h1[i];
    __syncthreads();
    out_h2[i] = gru_elem(i, hb, h1, w_ih_g, w_hh_g, b_ih_g, b_hh_g);
}

// ---------------------------------------------------------------------------
// Kernel 2: z = out_W @ h1 + out_b via V_WMMA_F32_16X16X4_F32.
// Each wave produces 16 rows; K=256 consumed in 64 chunks of 4.
// A 16x4 f32 layout: lane m=lane&15; v0 holds K=(lane<16?0:2), v1 holds K+1.
// B 4x16 uses the same per-lane K mapping; we broadcast h across all columns,
// so every column of D equals the GEMV result.
// ---------------------------------------------------------------------------
__global__ void logits_wmma(const float* __restrict__ W,
                            const float* __restrict__ bias,
                            const float* __restrict__ hvec,
                            float* __restrict__ z, int V) {
    __shared__ float hs[H];
    hs[threadIdx.x] = hvec[threadIdx.x];
    __syncthreads();

    const int gw = (blockIdx.x * blockDim.x + threadIdx.x) >> 5;  // wave id
    const int lane = threadIdx.x & 31;
    const int row0 = gw * 16;
    if (row0 >= V) return;  // wave-uniform: EXEC stays all-ones for WMMA

    const int m = lane & 15;
    const int kd = (lane < 16) ? 0 : 2;
    int row = row0 + m;
    if (row >= V) row = V - 1;  // clamp load addresses only
    const float* Wr = W + (size_t)row * H;

    v8f c = {0.f, 0.f, 0.f, 0.f, 0.f, 0.f, 0.f, 0.f};
#pragma unroll 4
    for (int k = 0; k < H; k += 4) {
        v2f a = *(const v2f*)(Wr + k + kd);   // global_load_b64
        v2f b = *(const v2f*)(hs + k + kd);   // ds_load_b64
        c = __builtin_amdgcn_wmma_f32_16x16x4_f32(
            false, a, false, b, (short)0, c, false, false);
    }

    // D layout: lane l, vgpr g -> row = row0 + g + (l>=16 ? 8 : 0), col = l&15.
    if ((lane & 15) == 0) {
        const int mbase = (lane >= 16) ? 8 : 0;
#pragma unroll
        for (int g = 0; g < 8; ++g) {
            int r = row0 + mbase + g;
            if (r < V) z[r] = c[g] + bias[r];
        }
    }
}

// ---------------------------------------------------------------------------
// Kernels 3-5: log-softmax over V = online (max, sum*exp) reduction.
// ---------------------------------------------------------------------------
__global__ void lse_partial(const float* __restrict__ z, int V,
                            float* __restrict__ part) {
    __shared__ float sm[256], ss[256];
    const int tid = threadIdx.x;
    float m = -3.4e38f, s = 0.f;
    for (int idx = blockIdx.x * blockDim.x + tid; idx < V;
         idx += gridDim.x * blockDim.x) {
        float v = z[idx];
        float M = fmaxf(m, v);
        s = s * expf(m - M) + expf(v - M);
        m = M;
    }
    sm[tid] = m; ss[tid] = s;
    __syncthreads();
    for (int st = 128; st > 0; st >>= 1) {
        if (tid < st) {
            float m1 = sm[tid], s1 = ss[tid];
            float m2 = sm[tid + st], s2 = ss[tid + st];
            float M = fmaxf(m1, m2);
            ss[tid] = s1 * expf(m1 - M) + s2 * expf(m2 - M);
            sm[tid] = M;
        }
        __syncthreads();
    }
    if (tid == 0) { part[2 * blockIdx.x] = sm[0]; part[2 * blockIdx.x + 1] = ss[0]; }
}

__global__ void lse_final(const float* __restrict__ part, float* __restrict__ lse) {
    __shared__ float sm[NPART], ss[NPART];
    const int i = threadIdx.x;
    sm[i] = part[2 * i]; ss[i] = part[2 * i + 1];
    __syncthreads();
    for (int st = NPART / 2; st > 0; st >>= 1) {
        if (i < st) {
            float m1 = sm[i], s1 = ss[i];
            float m2 = sm[i + st], s2 = ss[i + st];
            float M = fmaxf(m1, m2);
            ss[i] = s1 * expf(m1 - M) + s2 * expf(m2 - M);
            sm[i] = M;
        }
        __syncthreads();
    }
    if (i == 0) lse[0] = sm[0] + logf(ss[0]);
}

__global__ void logp_write(const float* __restrict__ z, const float* __restrict__ lse,
                           float* __restrict__ out, int V) {
    int i = blockIdx.x * blockDim.x + threadIdx.x;
    if (i < V) out[i] = z[i] - lse[0];
}

// ---------------------------------------------------------------------------
extern "C" void kernel_launch(void* const* d_in, const int* in_sizes, int n_in,
                              void* d_out, int out_size, void* d_ws, size_t ws_size,
                              hipStream_t stream) {
    const int*   token   = (const int*)  d_in[0];
    const float* hidden  = (const float*)d_in[1];
    const float* enc     = (const float*)d_in[2];
    const float* emb     = (const float*)d_in[3];
    const float* attn_W  = (const float*)d_in[4];
    const float* attn_b  = (const float*)d_in[5];
    const float* comb_W  = (const float*)d_in[6];
    const float* comb_b  = (const float*)d_in[7];
    const float* w_ih_f  = (const float*)d_in[8];
    const float* w_hh_f  = (const float*)d_in[9];
    const float* b_ih_f  = (const float*)d_in[10];
    const float* b_hh_f  = (const float*)d_in[11];
    const float* w_ih_b  = (const float*)d_in[12];
    const float* w_hh_b  = (const float*)d_in[13];
    const float* b_ih_b  = (const float*)d_in[14];
    const float* b_hh_b  = (const float*)d_in[15];
    const float* w_ih_g  = (const float*)d_in[16];
    const float* w_hh_g  = (const float*)d_in[17];
    const float* b_ih_g  = (const float*)d_in[18];
    const float* b_hh_g  = (const float*)d_in[19];
    const float* out_W   = (const float*)d_in[20];
    const float* out_b   = (const float*)d_in[21];

    const int V = in_sizes[21] > 0 ? in_sizes[21] : VOCAB_FALLBACK;

    float* out   = (float*)d_out;
    float* o_logp = out;            // [V]
    float* o_h2   = out + V;        // [H]
    float* o_aw   = out + V + H;    // [LSEQ]

    float* ws   = (float*)d_ws;
    float* wh1  = ws;               // [H]
    float* wz   = ws + H;           // [V]
    float* wpart = ws + H + V;      // [2*NPART]
    float* wlse  = wpart + 2 * NPART; // [1]

    decoder_small<<<1, H, 0, stream>>>(
        token, hidden, enc, emb, attn_W, attn_b, comb_W, comb_b,
        w_ih_f, w_hh_f, b_ih_f, b_hh_f, w_ih_b, w_hh_b, b_ih_b, b_hh_b,
        w_ih_g, w_hh_g, b_ih_g, b_hh_g, o_aw, o_h2, wh1);

    const int tiles = (V + 15) / 16;          // 16 rows per wave
    const int wblocks = (tiles + 7) / 8;      // 8 waves per 256-thread block
    logits_wmma<<<wblocks, 256, 0, stream>>>(out_W, out_b, wh1, wz, V);

    lse_partial<<<NPART, 256, 0, stream>>>(wz, V, wpart);
    lse_final<<<1, NPART, 0, stream>>>(wpart, wlse);
    logp_write<<<(V + 255) / 256, 256, 0, stream>>>(wz, wlse, o_logp, V);
}